// GCN_7516192768198
// MI455X (gfx1250) — compile-verified
//
#include <hip/hip_runtime.h>
#include <hip/hip_bf16.h>

#define N_NODES 50000
#define N_EDGES 800000

typedef float v2f __attribute__((ext_vector_type(2)));
typedef float v8f __attribute__((ext_vector_type(8)));

// ---------------------------------------------------------------- transpose
__global__ void gcn_transpose_k(const float* __restrict__ in, float* __restrict__ out,
                                int rows, int cols) {
    int gid = blockIdx.x * blockDim.x + threadIdx.x;
    if (gid < rows * cols) {
        int r = gid / cols;
        int c = gid - r * cols;
        out[(size_t)c * rows + r] = in[gid];   // out is [cols, rows]
    }
}

// ---------------------------------------------------- GEMM1: [N,300]@[300,256]
// one wave32 per 16x16 C tile; K stepped by 4 with V_WMMA_F32_16X16X4_F32
__global__ void gcn_gemm1_wmma(const float* __restrict__ x,
                               const float* __restrict__ w1t,
                               float* __restrict__ out) {
    const int wave    = (int)((blockIdx.x * blockDim.x + threadIdx.x) >> 5);
    const int lane    = threadIdx.x & 31;
    const int rowBase = (wave >> 4) << 4;   // 3125 row tiles
    const int colBase = (wave & 15) << 4;   // 16 col tiles
    const int m       = lane & 15;
    const int kl      = (lane >> 4) << 1;   // lanes 0-15 -> K+{0,1}; lanes 16-31 -> K+{2,3}

    const float* __restrict__ arow = x + (size_t)(rowBase + m) * 300;
    const float* __restrict__ bcol = w1t + colBase + m;

    v8f c = {0.f, 0.f, 0.f, 0.f, 0.f, 0.f, 0.f, 0.f};
    for (int k = 0; k < 300; k += 4) {
        v2f a = *(const v2f*)(arow + k + kl);
        v2f b;
        b.x = bcol[(size_t)(k + kl)     * 256];
        b.y = bcol[(size_t)(k + kl + 1) * 256];
        c = __builtin_amdgcn_wmma_f32_16x16x4_f32(false, a, false, b,
                                                  (short)0, c, false, false);
    }
    // C layout: VGPR j -> M = j (lanes 0-15) / j+8 (lanes 16-31), N = lane&15
    const int mBase = rowBase + ((lane >> 4) << 3);
    float* __restrict__ o = out + (size_t)mBase * 256 + colBase + m;
#pragma unroll
    for (int j = 0; j < 8; ++j)
        o[(size_t)j * 256] = c[j];
}

// ------------------------------------- GEMM2: relu(h1+b1)[N,256]@[256,128]
__global__ void gcn_gemm2_wmma(const float* __restrict__ h1,
                               const float* __restrict__ b1,
                               const float* __restrict__ w2t,
                               float* __restrict__ out) {
    const int wave    = (int)((blockIdx.x * blockDim.x + threadIdx.x) >> 5);
    const int lane    = threadIdx.x & 31;
    const int rowBase = (wave >> 3) << 4;   // 3125 row tiles
    const int colBase = (wave & 7)  << 4;   // 8 col tiles
    const int m       = lane & 15;
    const int kl      = (lane >> 4) << 1;

    const float* __restrict__ arow = h1 + (size_t)(rowBase + m) * 256;
    const float* __restrict__ bcol = w2t + colBase + m;

    v8f c = {0.f, 0.f, 0.f, 0.f, 0.f, 0.f, 0.f, 0.f};
    for (int k = 0; k < 256; k += 4) {
        v2f a  = *(const v2f*)(arow + k + kl);
        v2f bb = *(const v2f*)(b1   + k + kl);
        a.x = fmaxf(a.x + bb.x, 0.f);       // fused +bias, ReLU
        a.y = fmaxf(a.y + bb.y, 0.f);
        v2f b;
        b.x = bcol[(size_t)(k + kl)     * 128];
        b.y = bcol[(size_t)(k + kl + 1) * 128];
        c = __builtin_amdgcn_wmma_f32_16x16x4_f32(false, a, false, b,
                                                  (short)0, c, false, false);
    }
    const int mBase = rowBase + ((lane >> 4) << 3);
    float* __restrict__ o = out + (size_t)mBase * 128 + colBase + m;
#pragma unroll
    for (int j = 0; j < 8; ++j)
        o[(size_t)j * 128] = c[j];
}

// -------------------------------------------------- SpMM scatter via atomics
// one thread per (edge, 4-float chunk); dense operand is L2-resident (51 MB)
template <int D>
__global__ void gcn_spmm_scatter(const int*   __restrict__ erow,
                                 const int*   __restrict__ ecol,
                                 const float* __restrict__ eval,
                                 const float* __restrict__ dense,
                                 float*       __restrict__ out) {
    constexpr int CH = D / 4;                 // chunks per edge (64 or 32)
    constexpr int SH = (D == 256) ? 6 : 5;
    long gid = (long)blockIdx.x * blockDim.x + threadIdx.x;
    int  e   = (int)(gid >> SH);
    if (e >= N_EDGES) return;
    int  c   = ((int)gid & (CH - 1)) << 2;

    int   r  = erow[e];
    int   cc = ecol[e];
    float v  = eval[e];

    const float4 d = *(const float4*)(dense + (size_t)cc * D + c);
    float* o = out + (size_t)r * D + c;
    (void)__hip_atomic_fetch_add(o + 0, v * d.x, __ATOMIC_RELAXED, __HIP_MEMORY_SCOPE_AGENT);
    (void)__hip_atomic_fetch_add(o + 1, v * d.y, __ATOMIC_RELAXED, __HIP_MEMORY_SCOPE_AGENT);
    (void)__hip_atomic_fetch_add(o + 2, v * d.z, __ATOMIC_RELAXED, __HIP_MEMORY_SCOPE_AGENT);
    (void)__hip_atomic_fetch_add(o + 3, v * d.w, __ATOMIC_RELAXED, __HIP_MEMORY_SCOPE_AGENT);
}

// ------------------------------------------- +b2, row-wise L2-normalize (D=128)
__global__ void gcn_bias_norm(const float* __restrict__ b2, float* __restrict__ out) {
    int gid  = blockIdx.x * blockDim.x + threadIdx.x;
    int row  = gid >> 5;                      // one wave32 per row
    int lane = gid & 31;
    if (row >= N_NODES) return;

    float* p = out + (size_t)row * 128 + lane * 4;
    float4 v  = *(float4*)p;
    float4 bb = *(const float4*)(b2 + lane * 4);
    v.x += bb.x; v.y += bb.y; v.z += bb.z; v.w += bb.w;

    float ss = v.x * v.x + v.y * v.y + v.z * v.z + v.w * v.w;
#pragma unroll
    for (int off = 16; off > 0; off >>= 1)
        ss += __shfl_xor(ss, off, 32);

    float scale = 1.0f / fmaxf(sqrtf(ss), 1e-12f);
    v.x *= scale; v.y *= scale; v.z *= scale; v.w *= scale;
    *(float4*)p = v;
}

// ---------------------------------------------------------------- launcher
extern "C" void kernel_launch(void* const* d_in, const int* in_sizes, int n_in,
                              void* d_out, int out_size, void* d_ws, size_t ws_size,
                              hipStream_t stream) {
    const float* x    = (const float*)d_in[0];   // [50000,300]
    const int*   erow = (const int*)  d_in[1];   // [800000]
    const int*   ecol = (const int*)  d_in[2];   // [800000]
    const float* ev   = (const float*)d_in[3];   // [800000]
    const float* W1   = (const float*)d_in[4];   // [256,300]
    const float* b1   = (const float*)d_in[5];   // [256]
    const float* W2   = (const float*)d_in[6];   // [128,256]
    const float* b2   = (const float*)d_in[7];   // [128]
    float* out = (float*)d_out;                  // [50000,128]

    // workspace layout
    char* ws = (char*)d_ws;
    float* w1t  = (float*)(ws);                               // 300*256*4 = 307200 B
    float* w2t  = (float*)(ws + 307200);                      // 256*128*4 = 131072 B
    float* bufA = (float*)(ws + 307200 + 131072);             // 50000*256*4 = 51.2 MB
    float* bufB = (float*)(ws + 307200 + 131072 + 51200000);  // 50000*256*4 = 51.2 MB

    // 1. transpose weights (tiny)
    gcn_transpose_k<<<(256 * 300 + 255) / 256, 256, 0, stream>>>(W1, w1t, 256, 300);
    gcn_transpose_k<<<(128 * 256 + 255) / 256, 256, 0, stream>>>(W2, w2t, 128, 256);

    // 2. GEMM1: bufA = x @ W1.T   (3125*16 = 50000 waves, 8 waves/block)
    gcn_gemm1_wmma<<<50000 / 8, 256, 0, stream>>>(x, w1t, bufA);

    // 3. SpMM1: bufB = A @ bufA   (zero then scatter)
    hipMemsetAsync(bufB, 0, (size_t)N_NODES * 256 * sizeof(float), stream);
    gcn_spmm_scatter<256><<<(N_EDGES * 64) / 256, 256, 0, stream>>>(erow, ecol, ev, bufA, bufB);

    // 4. GEMM2: bufA = relu(bufB + b1) @ W2.T   (3125*8 = 25000 waves)
    gcn_gemm2_wmma<<<25000 / 8, 256, 0, stream>>>(bufB, b1, w2t, bufA);

    // 5. SpMM2: d_out = A @ bufA  (d_out is exactly [50000,128] f32)
    hipMemsetAsync(out, 0, (size_t)N_NODES * 128 * sizeof(float), stream);
    gcn_spmm_scatter<128><<<(N_EDGES * 32) / 256, 256, 0, stream>>>(erow, ecol, ev, bufA, out);

    // 6. +b2, row L2-normalize in place (one wave per row)
    gcn_bias_norm<<<(N_NODES * 32) / 256, 256, 0, stream>>>(b2, out);
}